// AvgNeighbor_74088185856029
// MI455X (gfx1250) — compile-verified
//
#include <hip/hip_runtime.h>

// ---------------------------------------------------------------------------
// COO SpMM:  out[r,:] += vals[e] * x[cols[e],:]   for e with rows[e]==r
// rows are sorted. N=50000, E=1250000, D=64. Output float32 [1,N,64].
// Wave-per-chunk segmented reduction, async global->LDS staging of edge
// metadata (double buffered via ASYNCcnt), atomic f32 flush at segment
// boundaries. x (12.8 MB) is L2-resident; gather traffic is served from L2.
// NOTE: spmm_coo_kernel is defined FIRST so the disasm snippet shows it.
// ---------------------------------------------------------------------------

#define D_FEAT 64

constexpr int kWaveSize       = 32;
constexpr int kWavesPerWG     = 8;
constexpr int kWGSize         = kWavesPerWG * kWaveSize;   // 256 threads
constexpr int kEdgesPerWave   = 128;                       // chunk per wave
constexpr int kBatch          = 32;                        // edges staged per step

#if defined(__has_builtin)
#if __has_builtin(__builtin_amdgcn_global_load_async_to_lds_b32) && \
    __has_builtin(__builtin_amdgcn_s_wait_asynccnt)
#define SPMM_USE_ASYNC_LDS 1
#endif
#endif

#ifndef SPMM_USE_ASYNC_LDS
#define SPMM_USE_ASYNC_LDS 0
#endif

#if SPMM_USE_ASYNC_LDS
// Builtin prototype (per hipcc diagnostic): (as1 int* src, as3 int* dst, imm, imm)
typedef __attribute__((address_space(1))) int gint_t;
typedef __attribute__((address_space(3))) int lint_t;
#define ASYNC_CP_B32(gsrc, ldst) \
  __builtin_amdgcn_global_load_async_to_lds_b32((gint_t*)(gsrc), (lint_t*)(ldst), 0, 0)
#define WAIT_ASYNC(n)                    \
  do {                                   \
    __builtin_amdgcn_s_wait_asynccnt(n); \
    asm volatile("" ::: "memory");       \
  } while (0)
#endif

// ------------------------------ SpMM ---------------------------------------
__global__ void __launch_bounds__(kWGSize)
spmm_coo_kernel(const float* __restrict__ x,     // [N, 64] (L2-resident, 12.8 MB)
                const float* __restrict__ vals,  // [E]
                const int*   __restrict__ rows,  // [E] sorted
                const int*   __restrict__ cols,  // [E]
                float* __restrict__ out,         // [N, 64], pre-zeroed
                int n_edges) {
  const int lane       = threadIdx.x & (kWaveSize - 1);
  const int wave_in_wg = threadIdx.x >> 5;
  const int wave_id    = blockIdx.x * kWavesPerWG + wave_in_wg;

  const int chunk_start = wave_id * kEdgesPerWave;
  if (chunk_start >= n_edges) return;
  const int chunk_end = min(chunk_start + kEdgesPerWave, n_edges);

  const float2* __restrict__ x2 = (const float2*)x;

  float2 acc = make_float2(0.f, 0.f);
  int cur_row = -1;

  auto flush = [&]() {
    if (cur_row >= 0) {
      atomicAdd(out + cur_row * D_FEAT + 2 * lane,     acc.x);
      atomicAdd(out + cur_row * D_FEAT + 2 * lane + 1, acc.y);
    }
  };
  auto do_edge = [&](int r, int c, float v) {
    if (r != cur_row) {          // wave-uniform branch (r is broadcast)
      flush();
      cur_row = r;
      acc.x = 0.f;
      acc.y = 0.f;
    }
    const float2 xv = x2[c * (D_FEAT / 2) + lane];   // coalesced 256B/wave, L2 hit
    acc.x = fmaf(v, xv.x, acc.x);
    acc.y = fmaf(v, xv.y, acc.y);
  };

#if SPMM_USE_ASYNC_LDS
  // Per-wave double-buffered staging: one int4 per edge {row, col, val, pad}
  // so the broadcast read-back is a single ds_load_b128.
  __shared__ int4 s_edge[kWavesPerWG][2][kBatch];

  auto stage = [&](int buf_idx, int base) {
    const int e = base + lane;
    if (e < chunk_end) {
      int4* dst = &s_edge[wave_in_wg][buf_idx][lane];
      ASYNC_CP_B32(rows + e, &dst->x);
      ASYNC_CP_B32(cols + e, &dst->y);
      ASYNC_CP_B32(vals + e, &dst->z);
    }
  };

  int buf = 0;
  stage(0, chunk_start);                       // ASYNCcnt = 3
  for (int b = chunk_start; b < chunk_end; b += kBatch) {
    const int nb = min(kBatch, chunk_end - b);
    const bool has_next = (b + kBatch) < chunk_end;
    if (has_next) {
      stage(buf ^ 1, b + kBatch);              // ASYNCcnt <= 6
      WAIT_ASYNC(3);                           // current buffer complete (in-order)
    } else {
      WAIT_ASYNC(0);
    }
    for (int j = 0; j < nb; ++j) {
      const int4 pk = s_edge[wave_in_wg][buf][j];   // ds_load_b128 broadcast
      do_edge(pk.x, pk.y, __int_as_float(pk.z));
    }
    buf ^= 1;
  }
#else
  // Fallback: cooperative register staging + wave shuffle broadcast.
  for (int b = chunk_start; b < chunk_end; b += kBatch) {
    const int nb = min(kBatch, chunk_end - b);
    int lr = 0, lc = 0;
    float lv = 0.f;
    const int e = b + lane;
    if (e < chunk_end) {
      lr = __builtin_nontemporal_load(rows + e);
      lc = __builtin_nontemporal_load(cols + e);
      lv = __builtin_nontemporal_load(vals + e);
    }
    for (int j = 0; j < nb; ++j) {
      const int   r = __shfl(lr, j, kWaveSize);
      const int   c = __shfl(lc, j, kWaveSize);
      const float v = __shfl(lv, j, kWaveSize);
      do_edge(r, c, v);
    }
  }
#endif

  flush();
}

// -------------------------- zero-init d_out --------------------------------
__global__ void __launch_bounds__(kWGSize)
spmm_zero_kernel(float* __restrict__ out, int n_total) {
  const int n4 = n_total >> 2;
  float4* __restrict__ out4 = (float4*)out;
  int i = blockIdx.x * blockDim.x + threadIdx.x;
  const int stride = gridDim.x * blockDim.x;
  for (; i < n4; i += stride) out4[i] = make_float4(0.f, 0.f, 0.f, 0.f);
  // scalar tail (none for D=64, kept for generality)
  if (blockIdx.x == 0 && threadIdx.x < (n_total & 3)) out[(n4 << 2) + threadIdx.x] = 0.f;
}

// ------------------------------ launch -------------------------------------
extern "C" void kernel_launch(void* const* d_in, const int* in_sizes, int n_in,
                              void* d_out, int out_size, void* d_ws, size_t ws_size,
                              hipStream_t stream) {
  const float* seq  = (const float*)d_in[0];   // [1, N, 64] float32
  const float* vals = (const float*)d_in[1];   // [E] float32
  const int*   rows = (const int*)d_in[2];     // [E] int32 (sorted)
  const int*   cols = (const int*)d_in[3];     // [E] int32
  float*       out  = (float*)d_out;           // [1, N, 64] float32

  const int n_edges = in_sizes[1];

  // 1) zero the output (atomic accumulation target; d_out is poisoned).
  {
    const int n4 = out_size >> 2;
    int blocks = (n4 + kWGSize - 1) / kWGSize;
    if (blocks < 1) blocks = 1;
    spmm_zero_kernel<<<blocks, kWGSize, 0, stream>>>(out, out_size);
  }

  // 2) segmented gather-scale-scatter.
  {
    const int waves  = (n_edges + kEdgesPerWave - 1) / kEdgesPerWave;
    const int blocks = (waves + kWavesPerWG - 1) / kWavesPerWG;
    spmm_coo_kernel<<<blocks, kWGSize, 0, stream>>>(seq, vals, rows, cols, out, n_edges);
  }
}